// SimpleGCN_1494648619174
// MI455X (gfx1250) — compile-verified
//
#include <hip/hip_runtime.h>

typedef __attribute__((ext_vector_type(2))) float v2f;
typedef __attribute__((ext_vector_type(8))) float v8f;

#define N_NODES 25000
#define N_EDGES 50000
#define N_GRAPHS 1000
#define IN_CH 9
#define HID 64
#define EDGE_DIM 3

#define EBLOCKS (N_EDGES / 16)          // 3125 exact
#define MBLOCKS ((N_NODES + 15) / 16)   // 1563 (last block partially valid)

__device__ __forceinline__ void atomic_add_f32(float* p, float v) {
    __hip_atomic_fetch_add(p, v, __ATOMIC_RELAXED, __HIP_MEMORY_SCOPE_AGENT);
}

// ---------------- h0 = relu(x @ W_emb + b_emb) : K=9, trivial VALU kernel ----
__global__ void k_embed(const float* __restrict__ x, const float* __restrict__ W,
                        const float* __restrict__ b, float* __restrict__ h) {
    int idx = blockIdx.x * blockDim.x + threadIdx.x;
    if (idx >= N_NODES * HID) return;
    int n = idx >> 6, o = idx & 63;
    float acc = b[o];
#pragma unroll
    for (int i = 0; i < IN_CH; ++i) acc += x[n * IN_CH + i] * W[i * HID + o];
    h[idx] = fmaxf(acc, 0.0f);
}

__global__ void k_zero(float* __restrict__ p, int n) {
    int i = blockIdx.x * blockDim.x + threadIdx.x;
    if (i < n) p[i] = 0.0f;
}

// ---------------- edge messages: agg[dst] += Σ_m scale_m * (h[src] @ W_m) ----
// 4 waves per block; wave w owns N-tile [16w, 16w+16). B fragments (4 matrices
// x 16 K-blocks x v2f = 128 VGPRs) stay register-resident across edge blocks.
__global__ void __launch_bounds__(128) k_msg(
        const float* __restrict__ h, const int* __restrict__ src,
        const int* __restrict__ dst, const float* __restrict__ ea,
        const float* __restrict__ We, const float* __restrict__ be,
        float* __restrict__ agg) {
    const int lane  = threadIdx.x & 31;
    const int wv    = threadIdx.x >> 5;
    const int n0    = wv * 16;
    const int lhalf = lane >> 4;   // 0 for lanes 0-15, 1 for lanes 16-31
    const int lmod  = lane & 15;

    // Load B fragments once: B[m][k] covers rows 4k..4k+3 of W_m, cols n0..n0+15.
    // Wave32 B layout: lane holds col n=lmod, K elems {4k+2*lhalf, 4k+2*lhalf+1}.
    v2f B[4][16];
#pragma unroll
    for (int m = 0; m < 4; ++m) {
        const float* Wm = (m < 3) ? (We + m * (HID * HID)) : be;
#pragma unroll
        for (int k = 0; k < 16; ++k) {
            int row = 4 * k + 2 * lhalf;
            B[m][k].x = Wm[row * HID + n0 + lmod];
            B[m][k].y = Wm[(row + 1) * HID + n0 + lmod];
        }
    }

    for (int eb = blockIdx.x; eb < EBLOCKS; eb += gridDim.x) {
        const int e_row = eb * 16 + lmod;       // A row owned by this lane
        const int srow  = src[e_row];           // fused gather
        const float* hrow = h + (long)srow * HID;

        v8f C0 = {}, C1 = {}, C2 = {}, C3 = {};
#pragma unroll
        for (int k = 0; k < 16; ++k) {
            // A layout: lane holds row lmod, K elems {4k+2*lhalf, 4k+2*lhalf+1}
            v2f A = *(const v2f*)(hrow + 4 * k + 2 * lhalf);
            C0 = __builtin_amdgcn_wmma_f32_16x16x4_f32(false, A, false, B[0][k],
                                                       (short)0, C0, false, false);
            C1 = __builtin_amdgcn_wmma_f32_16x16x4_f32(false, A, false, B[1][k],
                                                       (short)0, C1, false, false);
            C2 = __builtin_amdgcn_wmma_f32_16x16x4_f32(false, A, false, B[2][k],
                                                       (short)0, C2, false, false);
            C3 = __builtin_amdgcn_wmma_f32_16x16x4_f32(false, A, false, B[3][k],
                                                       (short)0, C3, false, false);
        }
        // C/D layout: lane l, VGPR v -> row M = v + 8*lhalf, col N = lmod.
#pragma unroll
        for (int v = 0; v < 8; ++v) {
            int e = eb * 16 + v + 8 * lhalf;
            float a0 = ea[e * EDGE_DIM + 0];
            float a1 = ea[e * EDGE_DIM + 1];
            float a2 = ea[e * EDGE_DIM + 2];
            float m  = a0 * C0[v] + a1 * C1[v] + a2 * C2[v] + C3[v];
            atomic_add_f32(&agg[(long)dst[e] * HID + n0 + lmod], m);
        }
    }
}

// ---------------- h_out = relu(agg + h @ root + bias) -----------------------
__global__ void __launch_bounds__(128) k_root(
        const float* __restrict__ h, const float* __restrict__ root,
        const float* __restrict__ bias, const float* __restrict__ agg,
        float* __restrict__ hout) {
    const int lane  = threadIdx.x & 31;
    const int wv    = threadIdx.x >> 5;
    const int n0    = wv * 16;
    const int lhalf = lane >> 4;
    const int lmod  = lane & 15;

    v2f B[16];
#pragma unroll
    for (int k = 0; k < 16; ++k) {
        int row = 4 * k + 2 * lhalf;
        B[k].x = root[row * HID + n0 + lmod];
        B[k].y = root[(row + 1) * HID + n0 + lmod];
    }
    const float bval = bias[n0 + lmod];

    for (int mb = blockIdx.x; mb < MBLOCKS; mb += gridDim.x) {
        int node   = mb * 16 + lmod;
        int nclamp = node < N_NODES ? node : (N_NODES - 1);  // pad rows: harmless
        const float* hrow = h + (long)nclamp * HID;

        v8f C = {};
#pragma unroll
        for (int k = 0; k < 16; ++k) {
            v2f A = *(const v2f*)(hrow + 4 * k + 2 * lhalf);
            C = __builtin_amdgcn_wmma_f32_16x16x4_f32(false, A, false, B[k],
                                                      (short)0, C, false, false);
        }
#pragma unroll
        for (int v = 0; v < 8; ++v) {
            int nd = mb * 16 + v + 8 * lhalf;
            if (nd < N_NODES) {
                long off  = (long)nd * HID + n0 + lmod;
                float val = C[v] + agg[off] + bval;
                hout[off] = fmaxf(val, 0.0f);
            }
        }
    }
}

// ---------------- pooled[batch[n]] += h[n] ----------------------------------
__global__ void k_pool(const float* __restrict__ h, const int* __restrict__ batch,
                       float* __restrict__ pooled) {
    int idx = blockIdx.x * blockDim.x + threadIdx.x;
    if (idx >= N_NODES * HID) return;
    int n = idx >> 6, c = idx & 63;
    atomic_add_f32(&pooled[(long)batch[n] * HID + c], h[idx]);
}

// ---------------- out = pooled @ W_out + b_out ------------------------------
__global__ void k_head(const float* __restrict__ pooled, const float* __restrict__ Wo,
                       const float* __restrict__ bo, float* __restrict__ out) {
    int g = blockIdx.x * blockDim.x + threadIdx.x;
    if (g >= N_GRAPHS) return;
    float acc = bo[0];
#pragma unroll
    for (int c = 0; c < HID; ++c) acc += pooled[g * HID + c] * Wo[c];
    out[g] = acc;
}

extern "C" void kernel_launch(void* const* d_in, const int* in_sizes, int n_in,
                              void* d_out, int out_size, void* d_ws, size_t ws_size,
                              hipStream_t stream) {
    const float* x    = (const float*)d_in[0];
    const int*   ei   = (const int*)d_in[1];
    const int*   src  = ei;
    const int*   dst  = ei + N_EDGES;
    const float* ea   = (const float*)d_in[2];
    const int*   batch= (const int*)d_in[3];
    const float* Wemb = (const float*)d_in[4];
    const float* bemb = (const float*)d_in[5];
    const float* We[3]   = {(const float*)d_in[6],  (const float*)d_in[10], (const float*)d_in[14]};
    const float* be[3]   = {(const float*)d_in[7],  (const float*)d_in[11], (const float*)d_in[15]};
    const float* rt[3]   = {(const float*)d_in[8],  (const float*)d_in[12], (const float*)d_in[16]};
    const float* bi[3]   = {(const float*)d_in[9],  (const float*)d_in[13], (const float*)d_in[17]};
    const float* Wout = (const float*)d_in[18];
    const float* bout = (const float*)d_in[19];
    float* out = (float*)d_out;

    float* ws     = (float*)d_ws;
    float* h0     = ws;                       // 1.6M floats
    float* h1     = ws + 1600000;             // 1.6M floats
    float* agg    = ws + 3200000;             // 1.6M floats
    float* pooled = ws + 4800000;             // 64K floats

    const int NH = N_NODES * HID;

    k_embed<<<(NH + 255) / 256, 256, 0, stream>>>(x, Wemb, bemb, h0);

    float* hin = h0;
    float* hout = h1;
    for (int l = 0; l < 3; ++l) {
        k_zero<<<(NH + 255) / 256, 256, 0, stream>>>(agg, NH);
        k_msg<<<512, 128, 0, stream>>>(hin, src, dst, ea, We[l], be[l], agg);
        k_root<<<512, 128, 0, stream>>>(hin, rt[l], bi[l], agg, hout);
        float* t = hin; hin = hout; hout = t;
    }

    k_zero<<<(N_GRAPHS * HID + 255) / 256, 256, 0, stream>>>(pooled, N_GRAPHS * HID);
    k_pool<<<(NH + 255) / 256, 256, 0, stream>>>(hin, batch, pooled);
    k_head<<<(N_GRAPHS + 255) / 256, 256, 0, stream>>>(pooled, Wout, bout, out);
}